// Block_68015102099504
// MI455X (gfx1250) — compile-verified
//
#include <hip/hip_runtime.h>
#include <math.h>

#define D_MODEL 1024
#define N_HEADS 16
#define HEAD_DIM 64
#define D_HIDDEN 4096
#define SEQ 2048
#define BATCH 4
#define EPS_RMS 1e-6f

typedef __attribute__((ext_vector_type(2))) float v2f;
typedef __attribute__((ext_vector_type(8))) float v8f;

// D = A(16x4) * B(4x16) + C, full fp32 (matches reference accumulation class)
__device__ __forceinline__ v8f wmma4(v2f a, v2f b, v8f c) {
  return __builtin_amdgcn_wmma_f32_16x16x4_f32(false, a, false, b, (short)0, c,
                                               false, false);
}

// ---------------------------------------------------------------------------
// RMSNorm: one 256-thread block per row of 1024
// ---------------------------------------------------------------------------
__global__ __launch_bounds__(256) void rmsnorm_kernel(
    const float* __restrict__ x, const float* __restrict__ w,
    float* __restrict__ y) {
  const int row = blockIdx.x;
  const float4 xv = ((const float4*)(x + (size_t)row * D_MODEL))[threadIdx.x];
  float ss = xv.x * xv.x + xv.y * xv.y + xv.z * xv.z + xv.w * xv.w;
#pragma unroll
  for (int off = 16; off > 0; off >>= 1) ss += __shfl_xor(ss, off, 32);
  __shared__ float part[8];
  if ((threadIdx.x & 31) == 0) part[threadIdx.x >> 5] = ss;
  __syncthreads();
  float tot = 0.f;
#pragma unroll
  for (int i = 0; i < 8; ++i) tot += part[i];
  const float r = rsqrtf(tot * (1.0f / D_MODEL) + EPS_RMS);
  const float4 wv = ((const float4*)w)[threadIdx.x];
  float4 o;
  o.x = wv.x * xv.x * r;
  o.y = wv.y * xv.y * r;
  o.z = wv.z * xv.z * r;
  o.w = wv.w * xv.w * r;
  ((float4*)(y + (size_t)row * D_MODEL))[threadIdx.x] = o;
}

// ---------------------------------------------------------------------------
// C[M,N] = A[M,K] @ W[N,K]^T (+bias) (+residual). Block tile 128x64, BK=16.
// 8 waves (4x2); each wave: 2x2 tiles of 16x16 via V_WMMA_F32_16X16X4_F32.
// Register double-buffering: next K-chunk's global loads overlap compute.
// ---------------------------------------------------------------------------
template <bool BIAS, bool RES>
__global__ __launch_bounds__(256) void gemm_wt_kernel(
    const float* __restrict__ A, const float* __restrict__ W,
    const float* __restrict__ bias, const float* __restrict__ R,
    float* __restrict__ C, int M, int N, int K) {
  constexpr int BM = 128, BN = 64, BK = 16, PAD = 20;
  __shared__ float As[BM * PAD];
  __shared__ float Ws[BN * PAD];
  const int m0 = blockIdx.y * BM;
  const int n0 = blockIdx.x * BN;
  const int t = threadIdx.x;
  const int lane = t & 31;
  const int wid = t >> 5;
  const int wm = wid >> 1, wn = wid & 1;
  const int half = lane >> 4, l16 = lane & 15;
  const int cc = half * 2;
  const int lr = t >> 2;       // 0..63
  const int lc = (t & 3) * 4;  // 0,4,8,12

  const float* pa0 = A + (size_t)(m0 + lr) * K + lc;
  const float* pa1 = A + (size_t)(m0 + lr + 64) * K + lc;
  const float* pw0 = W + (size_t)(n0 + lr) * K + lc;

  float4 a0 = *(const float4*)(pa0);
  float4 a1 = *(const float4*)(pa1);
  float4 w0 = *(const float4*)(pw0);

  v8f acc[2][2] = {};
  for (int kk = 0; kk < K; kk += BK) {
    *(float4*)(As + lr * PAD + lc) = a0;
    *(float4*)(As + (lr + 64) * PAD + lc) = a1;
    *(float4*)(Ws + lr * PAD + lc) = w0;
    __syncthreads();
    if (kk + BK < K) {  // issue next chunk's loads under current compute
      a0 = *(const float4*)(pa0 + kk + BK);
      a1 = *(const float4*)(pa1 + kk + BK);
      w0 = *(const float4*)(pw0 + kk + BK);
    }
#pragma unroll
    for (int k0 = 0; k0 < BK; k0 += 4) {
      v2f af[2], bf[2];
#pragma unroll
      for (int mt = 0; mt < 2; ++mt)
        af[mt] = *(const v2f*)(As + (wm * 32 + mt * 16 + l16) * PAD + k0 + cc);
#pragma unroll
      for (int nt = 0; nt < 2; ++nt)
        bf[nt] = *(const v2f*)(Ws + (wn * 32 + nt * 16 + l16) * PAD + k0 + cc);
#pragma unroll
      for (int mt = 0; mt < 2; ++mt)
#pragma unroll
        for (int nt = 0; nt < 2; ++nt)
          acc[mt][nt] = wmma4(af[mt], bf[nt], acc[mt][nt]);
    }
    __syncthreads();
  }
#pragma unroll
  for (int mt = 0; mt < 2; ++mt)
#pragma unroll
    for (int nt = 0; nt < 2; ++nt)
#pragma unroll
      for (int i = 0; i < 8; ++i) {
        const int row = m0 + wm * 32 + mt * 16 + half * 8 + i;
        const int col = n0 + wn * 32 + nt * 16 + l16;
        float val = acc[mt][nt][i];
        if (BIAS) val += bias[col];
        if (RES) val += R[(size_t)row * N + col];
        C[(size_t)row * N + col] = val;
      }
}

// ---------------------------------------------------------------------------
// Fused SwiGLU: U = silu(A@Wg^T) * (A@Wh^T). Same tiling, two acc sets.
// ---------------------------------------------------------------------------
__global__ __launch_bounds__(256) void swiglu_gemm_kernel(
    const float* __restrict__ A, const float* __restrict__ Wg,
    const float* __restrict__ Wh, float* __restrict__ U, int M, int N, int K) {
  constexpr int BM = 128, BN = 64, BK = 16, PAD = 20;
  __shared__ float As[BM * PAD];
  __shared__ float Gs[BN * PAD];
  __shared__ float Hs[BN * PAD];
  const int m0 = blockIdx.y * BM;
  const int n0 = blockIdx.x * BN;
  const int t = threadIdx.x;
  const int lane = t & 31;
  const int wid = t >> 5;
  const int wm = wid >> 1, wn = wid & 1;
  const int half = lane >> 4, l16 = lane & 15;
  const int cc = half * 2;
  const int lr = t >> 2;
  const int lc = (t & 3) * 4;

  const float* pa0 = A + (size_t)(m0 + lr) * K + lc;
  const float* pa1 = A + (size_t)(m0 + lr + 64) * K + lc;
  const float* pg0 = Wg + (size_t)(n0 + lr) * K + lc;
  const float* ph0 = Wh + (size_t)(n0 + lr) * K + lc;

  float4 a0 = *(const float4*)(pa0);
  float4 a1 = *(const float4*)(pa1);
  float4 g0 = *(const float4*)(pg0);
  float4 h0 = *(const float4*)(ph0);

  v8f accg[2][2] = {};
  v8f acch[2][2] = {};
  for (int kk = 0; kk < K; kk += BK) {
    *(float4*)(As + lr * PAD + lc) = a0;
    *(float4*)(As + (lr + 64) * PAD + lc) = a1;
    *(float4*)(Gs + lr * PAD + lc) = g0;
    *(float4*)(Hs + lr * PAD + lc) = h0;
    __syncthreads();
    if (kk + BK < K) {
      a0 = *(const float4*)(pa0 + kk + BK);
      a1 = *(const float4*)(pa1 + kk + BK);
      g0 = *(const float4*)(pg0 + kk + BK);
      h0 = *(const float4*)(ph0 + kk + BK);
    }
#pragma unroll
    for (int k0 = 0; k0 < BK; k0 += 4) {
      v2f af[2], gf[2], hf[2];
#pragma unroll
      for (int mt = 0; mt < 2; ++mt)
        af[mt] = *(const v2f*)(As + (wm * 32 + mt * 16 + l16) * PAD + k0 + cc);
#pragma unroll
      for (int nt = 0; nt < 2; ++nt) {
        gf[nt] = *(const v2f*)(Gs + (wn * 32 + nt * 16 + l16) * PAD + k0 + cc);
        hf[nt] = *(const v2f*)(Hs + (wn * 32 + nt * 16 + l16) * PAD + k0 + cc);
      }
#pragma unroll
      for (int mt = 0; mt < 2; ++mt)
#pragma unroll
        for (int nt = 0; nt < 2; ++nt) {
          accg[mt][nt] = wmma4(af[mt], gf[nt], accg[mt][nt]);
          acch[mt][nt] = wmma4(af[mt], hf[nt], acch[mt][nt]);
        }
    }
    __syncthreads();
  }
#pragma unroll
  for (int mt = 0; mt < 2; ++mt)
#pragma unroll
    for (int nt = 0; nt < 2; ++nt)
#pragma unroll
      for (int i = 0; i < 8; ++i) {
        const int row = m0 + wm * 32 + mt * 16 + half * 8 + i;
        const int col = n0 + wn * 32 + nt * 16 + l16;
        const float g = accg[mt][nt][i];
        const float h = acch[mt][nt][i];
        U[(size_t)row * N + col] = (g / (1.0f + expf(-g))) * h;
      }
}

// ---------------------------------------------------------------------------
// Flash attention, fp32 WMMA. One wave per 16-query tile; 8 waves/block.
// q,k,v,o layout: [B, L, N_HEADS*HEAD_DIM] row-major.
// ---------------------------------------------------------------------------
__global__ __launch_bounds__(256) void flash_attn_kernel(
    const float* __restrict__ Q, const float* __restrict__ K,
    const float* __restrict__ V, float* __restrict__ O) {
  __shared__ float pbuf[8][16 * 18];  // per-wave P transpose scratch
  const int t = threadIdx.x;
  const int w = t >> 5;
  const int lane = t & 31;
  const int half = lane >> 4, l16 = lane & 15;
  const int cc = half * 2;
  const int bh = blockIdx.y;
  const int b = bh >> 4;
  const int h = bh & 15;
  const size_t base = (size_t)b * SEQ * D_MODEL + (size_t)h * HEAD_DIM;
  const float* qp = Q + base;
  const float* kp = K + base;
  const float* vp = V + base;
  float* op = O + base;
  const int q0 = (blockIdx.x * 8 + w) * 16;

  // Preload Q fragments (per-lane 16 x float2), pre-scaled by 1/sqrt(64)=0.125
  v2f qf[16];
  {
    const float* qrow = qp + (size_t)(q0 + l16) * D_MODEL;
#pragma unroll
    for (int kk = 0; kk < 16; ++kk) {
      v2f qv = *(const v2f*)(qrow + kk * 4 + cc);
      qf[kk] = qv * 0.125f;
    }
  }
  float mi[8], li[8];
#pragma unroll
  for (int i = 0; i < 8; ++i) {
    mi[i] = -3.0e38f;
    li[i] = 0.0f;
  }
  v8f accO[4] = {};
  float* pw = &pbuf[w][0];

  for (int kt = 0; kt < SEQ / 16; ++kt) {
    // Prefetch next K/V tiles (global_prefetch) while computing this one.
    if (kt + 1 < SEQ / 16) {
      const size_t nrow = (size_t)((kt + 1) * 16 + l16) * D_MODEL;
      __builtin_prefetch(kp + nrow, 0, 3);
      __builtin_prefetch(vp + nrow, 0, 3);
    }
    // S = Q @ K^T  (16x16, K-dim = 64 => 16 k-steps)
    v8f s = {};
    const float* krow = kp + (size_t)(kt * 16 + l16) * D_MODEL;
#pragma unroll
    for (int kk = 0; kk < 16; ++kk) {
      v2f kf = *(const v2f*)(krow + kk * 4 + cc);
      s = wmma4(qf[kk], kf, s);
    }
    // Online softmax (row reductions across 16-lane halves)
    float p[8], alpha[8];
#pragma unroll
    for (int i = 0; i < 8; ++i) {
      float rm = s[i];
#pragma unroll
      for (int off = 1; off < 16; off <<= 1)
        rm = fmaxf(rm, __shfl_xor(rm, off, 32));
      const float mnew = fmaxf(mi[i], rm);
      alpha[i] = expf(mi[i] - mnew);
      const float pv = expf(s[i] - mnew);
      float rs = pv;
#pragma unroll
      for (int off = 1; off < 16; off <<= 1) rs += __shfl_xor(rs, off, 32);
      li[i] = alpha[i] * li[i] + rs;
      mi[i] = mnew;
      p[i] = pv;
    }
#pragma unroll
    for (int nt = 0; nt < 4; ++nt)
#pragma unroll
      for (int i = 0; i < 8; ++i) accO[nt][i] *= alpha[i];

    // Transpose P (C-layout -> A-layout) via per-wave LDS scratch.
#pragma unroll
    for (int i = 0; i < 8; ++i) pw[(half * 8 + i) * 18 + l16] = p[i];
    __syncthreads();  // uniform across all 8 waves every iteration
    v2f pf[4];
#pragma unroll
    for (int kk = 0; kk < 4; ++kk)
      pf[kk] = *(const v2f*)(pw + l16 * 18 + kk * 4 + cc);

    // O += P @ V  (K-dim = 16 => 4 k-steps, 4 N-tiles of 16)
    const float* vbase = vp + (size_t)(kt * 16) * D_MODEL;
#pragma unroll
    for (int kk = 0; kk < 4; ++kk) {
      const float* vr = vbase + (size_t)(kk * 4 + cc) * D_MODEL;
#pragma unroll
      for (int nt = 0; nt < 4; ++nt) {
        v2f vf;
        vf.x = vr[nt * 16 + l16];
        vf.y = vr[D_MODEL + nt * 16 + l16];
        accO[nt] = wmma4(pf[kk], vf, accO[nt]);
      }
    }
    __syncthreads();
  }
  // Write normalized output
#pragma unroll
  for (int i = 0; i < 8; ++i) {
    const float inv = 1.0f / li[i];
    float* orow = op + (size_t)(q0 + half * 8 + i) * D_MODEL;
#pragma unroll
    for (int nt = 0; nt < 4; ++nt) orow[nt * 16 + l16] = accO[nt][i] * inv;
  }
}

// ---------------------------------------------------------------------------
extern "C" void kernel_launch(void* const* d_in, const int* in_sizes, int n_in,
                              void* d_out, int out_size, void* d_ws,
                              size_t ws_size, hipStream_t stream) {
  const float* x = (const float*)d_in[0];
  const float* Wq = (const float*)d_in[1];
  const float* Wk = (const float*)d_in[2];
  const float* Wv = (const float*)d_in[3];
  const float* Wo = (const float*)d_in[4];
  const float* bo = (const float*)d_in[5];
  const float* anw = (const float*)d_in[6];
  const float* fnw = (const float*)d_in[7];
  const float* Wg = (const float*)d_in[8];
  const float* Wh = (const float*)d_in[9];
  const float* Wout = (const float*)d_in[10];
  const float* bout = (const float*)d_in[11];
  float* out = (float*)d_out;
  float* ws = (float*)d_ws;

  const int TOK = BATCH * SEQ;  // 8192
  float* xn = ws;
  float* q = xn + (size_t)TOK * D_MODEL;
  float* k = q + (size_t)TOK * D_MODEL;
  float* v = k + (size_t)TOK * D_MODEL;
  float* ao = v + (size_t)TOK * D_MODEL;
  float* u = ao + (size_t)TOK * D_MODEL;  // [TOK, D_HIDDEN]

  dim3 blk(256);
  // 1) xn = rmsnorm(x, attn_norm_w)
  rmsnorm_kernel<<<TOK, blk, 0, stream>>>(x, anw, xn);
  // 2) q,k,v projections
  dim3 g1(D_MODEL / 64, TOK / 128);
  gemm_wt_kernel<false, false>
      <<<g1, blk, 0, stream>>>(xn, Wq, nullptr, nullptr, q, TOK, D_MODEL, D_MODEL);
  gemm_wt_kernel<false, false>
      <<<g1, blk, 0, stream>>>(xn, Wk, nullptr, nullptr, k, TOK, D_MODEL, D_MODEL);
  gemm_wt_kernel<false, false>
      <<<g1, blk, 0, stream>>>(xn, Wv, nullptr, nullptr, v, TOK, D_MODEL, D_MODEL);
  // 3) flash attention -> ao
  dim3 ga(SEQ / (16 * 8), BATCH * N_HEADS);
  flash_attn_kernel<<<ga, blk, 0, stream>>>(q, k, v, ao);
  // 4) x1 = x + ao @ Wo^T + bo   (stored in d_out)
  gemm_wt_kernel<true, true>
      <<<g1, blk, 0, stream>>>(ao, Wo, bo, x, out, TOK, D_MODEL, D_MODEL);
  // 5) xn = rmsnorm(x1, ffn_norm_w)
  rmsnorm_kernel<<<TOK, blk, 0, stream>>>(out, fnw, xn);
  // 6) u = silu(xn@Wg^T) * (xn@Wh^T)
  dim3 g2(D_HIDDEN / 64, TOK / 128);
  swiglu_gemm_kernel<<<g2, blk, 0, stream>>>(xn, Wg, Wh, u, TOK, D_HIDDEN, D_MODEL);
  // 7) out = x1 + u @ Wout^T + bout
  gemm_wt_kernel<true, true>
      <<<g1, blk, 0, stream>>>(u, Wout, bout, out, out, TOK, D_MODEL, D_HIDDEN);
}